// label_swapper_dynamic_71030169141884
// MI455X (gfx1250) — compile-verified
//
#include <hip/hip_runtime.h>
#include <hip/hip_bf16.h>

typedef __attribute__((ext_vector_type(2))) float v2f;
typedef __attribute__((ext_vector_type(8))) float v8f;

#define NTAB   50000
#define NQ     4096
#define NCLS   10
#define NTILES 3125   // 50000 / 16
#define MTILES 256    // 4096 / 16

// ---------------------------------------------------------------------------
// Pass 0: per-table-row squared norms (|t_j|^2)
// ---------------------------------------------------------------------------
__global__ __launch_bounds__(256) void tnorm_kernel(const float* __restrict__ tab,
                                                    float* __restrict__ tnorm) {
    int i = blockIdx.x * blockDim.x + threadIdx.x;
    if (i >= NTAB) return;
    const float* t = tab + i * NCLS;
    float s = 0.f;
#pragma unroll
    for (int k = 0; k < NCLS; ++k) s = fmaf(t[k], t[k], s);
    tnorm[i] = s;
}

// ---------------------------------------------------------------------------
// Pass 1: WMMA f32 16x16x4 GEMM + streaming argmin of (|t|^2 - 2 q.t)
// One wave handles one (16-query M-tile, table segment) pair.
// Branchless inner loop: A's K=10,11 pad lanes are zero, so B's pad lanes
// may hold arbitrary finite in-bounds data (0 * finite == 0 exactly).
// ---------------------------------------------------------------------------
__global__ __launch_bounds__(256) void argmin_kernel(const float* __restrict__ tab,
                                                     const float* __restrict__ qry,
                                                     const float* __restrict__ tnorm,
                                                     float* __restrict__ pval,
                                                     int* __restrict__ pidx,
                                                     int nseg) {
    const int lane  = threadIdx.x & 31;
    const int wid   = blockIdx.x * 8 + (threadIdx.x >> 5);
    const int mtile = wid & (MTILES - 1);
    const int seg   = wid >> 8;
    const int h     = lane >> 4;   // half-wave: selects K pair
    const int idx   = lane & 15;   // M row (A) / N column (B)

    // A-matrix (16 queries x K), loop invariant.
    // ISA layout 16x4 f32: lane<16 -> M=lane, K={2h,2h+1} per VGPR pair.
    const float* qp = qry + (mtile * 16 + idx) * NCLS;
    v2f a0 = *(const v2f*)(qp + 2 * h);        // K 0..3
    v2f a1 = *(const v2f*)(qp + 4 + 2 * h);    // K 4..7
    v2f a2;                                    // K 8..11 (10,11 MUST be zero)
    if (h == 0) a2 = *(const v2f*)(qp + 8);
    else        { a2.x = 0.f; a2.y = 0.f; }

    float mv[8];
    int   mi[8];
#pragma unroll
    for (int r = 0; r < 8; ++r) { mv[r] = 3.4e38f; mi[r] = 0x7fffffff; }

    const int per = (NTILES + nseg - 1) / nseg;
    int t0 = seg * per;
    int t1 = t0 + per;
    if (t1 > NTILES) t1 = NTILES;

    // Loop-invariant per-lane B offsets. For h=1 lanes, offb2 points at
    // finite in-bounds garbage (next row's k=0,1; clamped to own row for
    // idx==15) which is annihilated by the zero A pad lanes.
    const int offb0 = 2 * h;
    const int offb1 = 4 + 2 * h;
    const int offb2 = 8 + 2 * (h & (idx < 15 ? 1 : 0));

    const float* tp = tab + (t0 * 16 + idx) * NCLS;   // running row pointer
    for (int t = t0; t < t1; ++t, tp += 16 * NCLS) {
        const int j = t * 16 + idx;                    // this lane's table column
        v2f b0 = *(const v2f*)(tp + offb0);
        v2f b1 = *(const v2f*)(tp + offb1);
        v2f b2 = *(const v2f*)(tp + offb2);

        v8f c = {0.f, 0.f, 0.f, 0.f, 0.f, 0.f, 0.f, 0.f};
        c = __builtin_amdgcn_wmma_f32_16x16x4_f32(false, a0, false, b0, (short)0, c, false, false);
        c = __builtin_amdgcn_wmma_f32_16x16x4_f32(false, a1, false, b1, (short)0, c, false, false);
        c = __builtin_amdgcn_wmma_f32_16x16x4_f32(false, a2, false, b2, (short)0, c, false, false);

        const float tn = tnorm[j];
#pragma unroll
        for (int r = 0; r < 8; ++r) {
            float s = fmaf(-2.0f, c[r], tn);   // |t|^2 - 2 q.t  (|q|^2 drops out)
            if (s < mv[r]) { mv[r] = s; mi[r] = j; }
        }
    }

    // Cross-lane argmin over the 16 columns held by each half-wave.
    // D layout: lanes 0-15 -> M = r, lanes 16-31 -> M = r + 8.
#pragma unroll
    for (int r = 0; r < 8; ++r) {
        float v = mv[r];
        int   id = mi[r];
#pragma unroll
        for (int m = 1; m <= 8; m <<= 1) {
            float ov = __shfl_xor(v, m, 32);
            int   oi = __shfl_xor(id, m, 32);
            if (ov < v || (ov == v && oi < id)) { v = ov; id = oi; }
        }
        if (idx == 0) {
            int row = mtile * 16 + 8 * h + r;
            pval[row * nseg + seg] = v;
            pidx[row * nseg + seg] = id;
        }
    }
}

// ---------------------------------------------------------------------------
// Pass 2: reduce segment partials, exact d2 for winner, label-swap logic.
// ---------------------------------------------------------------------------
__global__ __launch_bounds__(256) void finalize_kernel(const float* __restrict__ tab,
                                                       const float* __restrict__ qry,
                                                       const int* __restrict__ flip_table,
                                                       const int* __restrict__ flip_offset,
                                                       const float* __restrict__ pval,
                                                       const int* __restrict__ pidx,
                                                       int nseg,
                                                       float* __restrict__ out,
                                                       int* __restrict__ count) {
    int i = blockIdx.x * blockDim.x + threadIdx.x;
    if (i >= NQ) return;

    float bv = 3.5e38f;
    int   bi = 0x7fffffff;
    for (int s = 0; s < nseg; ++s) {
        float v = pval[i * nseg + s];
        int   id = pidx[i * nseg + s];
        if (v < bv || (v == bv && id < bi)) { bv = v; bi = id; }
    }

    const float* q = qry + i * NCLS;
    const float* t = tab + bi * NCLS;

    // Exact distance for the winning candidate: 0 exactly for a true match.
    float d2 = 0.f;
#pragma unroll
    for (int k = 0; k < NCLS; ++k) { float d = q[k] - t[k]; d2 = fmaf(d, d, d2); }
    bool iskey = sqrtf(fmaxf(d2, 0.f)) <= 1e-5f;   // keys != -1

    // argmax(query row) with first-occurrence tie-break (strict >)
    int   tl = 0;
    float tv = q[0];
#pragma unroll
    for (int c = 1; c < NCLS; ++c) if (q[c] > tv) { tv = q[c]; tl = c; }

    int  off    = (iskey && flip_table[i] == 1) ? flip_offset[i] : 0;
    int  fl     = (tl + off) % NCLS;
    bool member = iskey && (flip_table[bi] == 1);

#pragma unroll
    for (int c = 0; c < NCLS; ++c) {
        int src = c;
        if (member) {
            if (c == tl)      src = fl;
            else if (c == fl) src = tl;
        }
        out[i * NCLS + c] = q[src];
    }

    if (member && (tl != fl)) atomicAdd(count, 1);
}

__global__ void write_num_kernel(const int* __restrict__ count, float* __restrict__ dst) {
    *dst = (float)(*count);
}

// ---------------------------------------------------------------------------
extern "C" void kernel_launch(void* const* d_in, const int* in_sizes, int n_in,
                              void* d_out, int out_size, void* d_ws, size_t ws_size,
                              hipStream_t stream) {
    const float* tab = (const float*)d_in[0];
    const float* qry = (const float*)d_in[1];
    const int*   ft  = (const int*)d_in[2];
    const int*   fo  = (const int*)d_in[3];
    float*       out = (float*)d_out;
    char*        ws  = (char*)d_ws;

    // Workspace layout: tnorm[50000] f32 | count i32 | pval[4096*nseg] | pidx[4096*nseg]
    float* tnorm = (float*)ws;                       // [0, 200000)
    int*   count = (int*)(ws + 200000);              // [200000, 200004)
    const size_t pbase = 200064;

    int nseg = 16;
    if (ws_size > pbase) {
        size_t maxseg = (ws_size - pbase) / ((size_t)NQ * 8);
        if ((size_t)nseg > maxseg) nseg = (int)maxseg;
    } else {
        nseg = 1;
    }
    if (nseg < 1)  nseg = 1;
    if (nseg > 16) nseg = 16;

    float* pval = (float*)(ws + pbase);
    int*   pidx = (int*)(ws + pbase + (size_t)NQ * nseg * sizeof(float));

    hipMemsetAsync(count, 0, sizeof(int), stream);

    tnorm_kernel<<<(NTAB + 255) / 256, 256, 0, stream>>>(tab, tnorm);
    argmin_kernel<<<32 * nseg, 256, 0, stream>>>(tab, qry, tnorm, pval, pidx, nseg);
    finalize_kernel<<<NQ / 256, 256, 0, stream>>>(tab, qry, ft, fo, pval, pidx, nseg, out, count);
    if (out_size > NQ * NCLS) {
        write_num_kernel<<<1, 1, 0, stream>>>(count, out + NQ * NCLS);
    }
}